// GCN_Layer_1683627180840
// MI455X (gfx1250) — compile-verified
//
#include <hip/hip_runtime.h>
#include <stdint.h>

#define F_DIM 128
#define WAVES 4      // waves per block, one row per wave
#define WTILE 128    // edges staged per wave per stage: 32 lanes * 4 edges (16B b128 DMA)

// ---------------------------------------------------------------- utilities
__global__ void k_zero_i32(int* __restrict__ p, int n) {
  int i = blockIdx.x * blockDim.x + threadIdx.x;
  if (i < n) p[i] = 0;
}

__global__ void k_count(const int* __restrict__ rows, int e, int* __restrict__ cnt) {
  int i = blockIdx.x * blockDim.x + threadIdx.x;
  if (i < e) atomicAdd(&cnt[rows[i]], 1);
}

// Single block of 1024 threads: exclusive scan cnt[0..n) -> row_ptr[0..n],
// and nxt[i] = row_ptr[i].  cnt and nxt may alias (read-before-write per index).
__global__ void k_scan(const int* cnt, int* row_ptr, int* nxt, int n) {
  __shared__ int part[1024];
  __shared__ int pscan[1025];
  const int t = threadIdx.x;
  const int chunk = (n + 1023) >> 10;
  const int b = t * chunk;
  const int e = min(b + chunk, n);
  int s = 0;
  for (int i = b; i < e; ++i) s += cnt[i];
  part[t] = s;
  __syncthreads();
  if (t == 0) {
    int acc = 0;
    for (int i = 0; i < 1024; ++i) { pscan[i] = acc; acc += part[i]; }
    pscan[1024] = acc;
  }
  __syncthreads();
  int acc = pscan[t];
  for (int i = b; i < e; ++i) {
    int c = cnt[i];          // read before write: cnt may alias nxt
    row_ptr[i] = acc;
    nxt[i]     = acc;
    acc += c;
  }
  if (t == 0) row_ptr[n] = pscan[1024];
}

__global__ void k_fill(const float* __restrict__ vals, const int* __restrict__ rows,
                       const int* __restrict__ cols, int e,
                       int* __restrict__ nxt, float* __restrict__ cval,
                       int* __restrict__ ccol) {
  int i = blockIdx.x * blockDim.x + threadIdx.x;
  if (i < e) {
    int pos = atomicAdd(&nxt[rows[i]], 1);
    cval[pos] = vals[i];
    ccol[pos] = cols[i];
  }
}

// ----------------------------------------------------------------- SpMM hop
// LDS byte-offset of a __shared__ object (AS3 pointers are 32-bit offsets).
template <typename T>
static __device__ __forceinline__ uint32_t lds_addr32(T* p) {
  return (uint32_t)(uintptr_t)(__attribute__((address_space(3))) T*)p;
}

// One wave32 per row. Each lane owns a 4-feature quad (float4): every gather of
// h[col] is one coalesced 512B b128 load per wave. Edge (val,col) tiles are
// staged into a PER-WAVE LDS region via async DMA (GLOBAL_LOAD_ASYNC_TO_LDS_B128,
// tracked by ASYNCcnt) -> wave-synchronous, zero block barriers, zero fp atomics.
__global__ void __launch_bounds__(32 * WAVES)
k_spmm(const float* __restrict__ h, const float* __restrict__ cval,
       const int* __restrict__ ccol, const int* __restrict__ row_ptr,
       float* __restrict__ out, int n) {
  __shared__ float sval[WAVES][WTILE];
  __shared__ int   scol[WAVES][WTILE];
  const int wv   = threadIdx.x >> 5;
  const int lane = threadIdx.x & 31;
  const int r    = blockIdx.x * WAVES + wv;
  if (r >= n) return;
  const int start = row_ptr[r];
  const int end   = row_ptr[r + 1];
  const uint32_t vbase = lds_addr32(&sval[wv][0]);
  const uint32_t cbase = lds_addr32(&scol[wv][0]);
  const int fq = lane * 4;                  // this lane's feature quad
  float4 acc = make_float4(0.f, 0.f, 0.f, 0.f);

  // Tiles snapped to 4-edge (16B) boundaries so b128 async DMA stays aligned.
  for (int wb = (start & ~3); wb < end; wb += WTILE) {
    const int lo   = max(start, wb);
    const int hi   = min(end, wb + WTILE);
    const int ngrp = (hi - wb + 3) >> 2;    // 16B groups to stage (<= 32)
    if (lane < ngrp) {
      const float* gv = cval + wb + 4 * lane;
      const int*   gc = ccol + wb + 4 * lane;
      asm volatile("global_load_async_to_lds_b128 %0, %1, off"
                   :: "v"(vbase + 16u * (uint32_t)lane), "v"(gv) : "memory");
      asm volatile("global_load_async_to_lds_b128 %0, %1, off"
                   :: "v"(cbase + 16u * (uint32_t)lane), "v"(gc) : "memory");
    }
    asm volatile("s_wait_asynccnt 0x0" ::: "memory");  // own wave's DMA done

    const int jend = hi - wb;
    #pragma unroll 4
    for (int j = lo - wb; j < jend; ++j) {
      const float  v  = sval[wv][j];        // LDS broadcast reads
      const float4 hr = *(const float4*)(h + (size_t)scol[wv][j] * F_DIM + fq);
      acc.x = fmaf(v, hr.x, acc.x);
      acc.y = fmaf(v, hr.y, acc.y);
      acc.z = fmaf(v, hr.z, acc.z);
      acc.w = fmaf(v, hr.w, acc.w);
    }
    // No barrier needed: per-wave tile; all ds_load results are consumed into
    // registers before the next (asm-ordered) stage overwrites the tile.
  }
  *(float4*)(out + (size_t)r * F_DIM + fq) = acc;
}

// -------------------------------------------------------------------- entry
extern "C" void kernel_launch(void* const* d_in, const int* in_sizes, int n_in,
                              void* d_out, int out_size, void* d_ws, size_t ws_size,
                              hipStream_t stream) {
  (void)n_in; (void)out_size; (void)ws_size;
  const float* x  = (const float*)d_in[0];
  const float* ev = (const float*)d_in[1];
  const int*   er = (const int*)d_in[2];
  const int*   ec = (const int*)d_in[3];
  const int NF = in_sizes[0];
  const int E  = in_sizes[1];
  const int N  = NF / F_DIM;
  const int K  = 3;   // K_HOPS is a static constant in the reference

  auto al = [](size_t v) { return (v + 255) & ~(size_t)255; };
  char* w = (char*)d_ws;
  int*   row_ptr = (int*)w;   w += al((size_t)(N + 1) * 4);
  int*   cntnxt  = (int*)w;   w += al((size_t)N * 4);
  float* cval    = (float*)w; w += al((size_t)E * 4);
  int*   ccol    = (int*)w;   w += al((size_t)E * 4);
  float* h1      = (float*)w; // NF floats; placed last so 12B staging
                              // overreads of ccol land harmlessly here

  // Build CSR (rebuilt every call; all state lives in ws).
  k_zero_i32<<<(N + 255) / 256, 256, 0, stream>>>(cntnxt, N);
  k_count  <<<(E + 255) / 256, 256, 0, stream>>>(er, E, cntnxt);
  k_scan   <<<1, 1024, 0, stream>>>(cntnxt, row_ptr, cntnxt, N);
  k_fill   <<<(E + 255) / 256, 256, 0, stream>>>(ev, er, ec, E, cntnxt, cval, ccol);

  // k = 3 hops, ping-pong so the final hop lands in d_out: x -> out -> h1 -> out
  float* out = (float*)d_out;
  const float* src = x;
  const int grid = (N + WAVES - 1) / WAVES;
  for (int hop = 0; hop < K; ++hop) {
    float* dst = (((K - hop) & 1) == 1) ? out : h1;
    k_spmm<<<grid, 32 * WAVES, 0, stream>>>(src, cval, ccol, row_ptr, dst, N);
    src = dst;
  }
}